// ODDLinear_36867999268894
// MI455X (gfx1250) — compile-verified
//
#include <hip/hip_runtime.h>
#include <stdint.h>

#define B_   64
#define T_   336
#define C_   1024
#define PADW 12            // (25-1)/2

#define PAIR_STRIDE 160    // LDS floats per t-pair row [B][2] + pad (conflict-free)
#define W_STRIDE    340    // LDS row stride for 16x336 weight tiles (conflict-free)
#define LDS_ACT  ((T_ / 2) * PAIR_STRIDE)  // floats per activation array
#define LDS_W    (16 * W_STRIDE)           // floats per weight tile (one matrix)

typedef __attribute__((ext_vector_type(2))) float v2f;
typedef __attribute__((ext_vector_type(8))) float v8f;

// CDNA5 async global->LDS copy (GV form: per-lane 64-bit address), ASYNCcnt-tracked.
__device__ __forceinline__ void async_b128(uint32_t lds_byte, const void* gaddr) {
  asm volatile("global_load_async_to_lds_b128 %0, %1, off"
               :: "v"(lds_byte), "v"((uint64_t)(uintptr_t)gaddr)
               : "memory");
}

// ---------------------------------------------------------------------------
// Pass 1: decomposition.  x:[B,T,C] fp32 -> seasonal/trend ws in K-paired
// layout [C][T/2][B][2].  Two time steps per iteration: coalesced float2
// global writes; stride-130 LDS tile rows are bank-conflict free both ways.
// ---------------------------------------------------------------------------
__global__ __launch_bounds__(256) void decompose_kernel(
    const float* __restrict__ x, float* __restrict__ s_ws, float* __restrict__ t_ws,
    int tslab) {
  __shared__ float sS[32 * 130];
  __shared__ float sT[32 * 130];
  const int cg  = blockIdx.x;        // channel group of 32
  const int t0  = blockIdx.y * tslab;
  const int t1  = t0 + tslab;
  const int tid = threadIdx.x;

  float run[8];
  int   base[8];
#pragma unroll
  for (int j = 0; j < 8; ++j) {
    const int p  = tid + 256 * j;
    const int cl = p & 31;           // channel-in-group: lane-consecutive reads
    const int b  = p >> 5;
    base[j] = (b * T_) * C_ + cg * 32 + cl;
    float s = 0.0f;
    for (int q = t0 - PADW; q <= t0 + PADW; ++q) {
      int qc = q < 0 ? 0 : (q > T_ - 1 ? T_ - 1 : q);
      s += x[base[j] + qc * C_];
    }
    run[j] = s;
  }

  for (int t = t0; t < t1; t += 2) {   // tslab is even
#pragma unroll
    for (int j = 0; j < 8; ++j) {
      const int p  = tid + 256 * j;
      const int cl = p & 31;
      const int b  = p >> 5;
      // step t
      const int tp0 = (t + PADW + 1 < T_) ? (t + PADW + 1) : (T_ - 1);
      const int tm0 = (t - PADW > 0) ? (t - PADW) : 0;
      const float xc0 = x[base[j] + t * C_];
      const float tr0 = run[j] * (1.0f / 25.0f);
      run[j] += x[base[j] + tp0 * C_] - x[base[j] + tm0 * C_];
      // step t+1
      const int tp1 = (t + PADW + 2 < T_) ? (t + PADW + 2) : (T_ - 1);
      const int tm1 = (t + 1 - PADW > 0) ? (t + 1 - PADW) : 0;
      const float xc1 = x[base[j] + (t + 1) * C_];
      const float tr1 = run[j] * (1.0f / 25.0f);
      run[j] += x[base[j] + tp1 * C_] - x[base[j] + tm1 * C_];

      v2f vs, vt;
      vs[0] = xc0 - tr0;  vs[1] = xc1 - tr1;
      vt[0] = tr0;        vt[1] = tr1;
      *(v2f*)(sS + cl * 130 + 2 * b) = vs;
      *(v2f*)(sT + cl * 130 + 2 * b) = vt;
    }
    __syncthreads();
#pragma unroll
    for (int j = 0; j < 8; ++j) {
      const int q  = tid + 256 * j;
      const int cw = q >> 6;
      const int bw = q & 63;
      const int g = (((cg * 32 + cw) * (T_ / 2) + (t >> 1)) * B_ + bw) * 2;
      *(v2f*)(s_ws + g) = *(const v2f*)(sS + cw * 130 + 2 * bw);
      *(v2f*)(t_ws + g) = *(const v2f*)(sT + cw * 130 + 2 * bw);
    }
    __syncthreads();
  }
}

// ---------------------------------------------------------------------------
// Pass 2: per-channel grouped GEMM via V_WMMA_F32_16X16X4_F32.
// Activations resident in LDS (async prologue, K-paired rows: one b64 per
// A fragment).  16x336 weight tiles double-buffered and streamed with
// GLOBAL_LOAD_ASYNC_TO_LDS_B128; DMA of tile nt+1 overlaps WMMA on tile nt.
// 21 async instrs/wave/tile -> s_wait_asynccnt 21 retires exactly tile nt.
// ---------------------------------------------------------------------------
__global__ __launch_bounds__(128) void odd_gemm_kernel(
    const float* __restrict__ s_act, const float* __restrict__ t_act,
    const float* __restrict__ Ws, const float* __restrict__ bs,
    const float* __restrict__ Wt, const float* __restrict__ bt,
    float* __restrict__ out) {
  extern __shared__ float lds[];
  float* aS = lds;
  float* aT = lds + LDS_ACT;

  const int c    = blockIdx.x;
  const int tid  = threadIdx.x;
  const int lane = tid & 31;
  const int wave = tid >> 5;        // b-tile index 0..3
  const int m    = lane & 15;
  const int h    = lane >> 4;       // K-half select per ISA A/B layout

  const uint32_t lds0    = (uint32_t)(uintptr_t)lds;   // wave-relative LDS bytes
  const uint32_t actS_b  = lds0;
  const uint32_t actT_b  = lds0 + (uint32_t)LDS_ACT * 4;
  const uint32_t wbuf_b0 = lds0 + (uint32_t)(2 * LDS_ACT) * 4;
  const uint32_t wbufSz  = (uint32_t)(2 * LDS_W) * 4;  // one slot (Ws+Wt)

  // ---- async prologue: activations [T/2][128] -> LDS rows of PAIR_STRIDE ----
  {
    const size_t cbase = (size_t)c * (T_ * B_);
    for (int i = 0; i < (2 * T_ * B_) / (4 * 128); ++i) {   // 84 iters, exact
      const int idx  = tid + 128 * i;                       // 0..10751
      const int matT = idx >= (T_ * B_ / 4);
      const int f    = matT ? idx - (T_ * B_ / 4) : idx;
      const int e    = f * 4;                               // float index
      const int r    = e >> 7;                              // t-pair row
      const int col  = e & 127;
      const float* src = (matT ? t_act : s_act) + cbase + e;
      const uint32_t dst = (matT ? actT_b : actS_b) +
                           (uint32_t)(r * PAIR_STRIDE + col) * 4;
      async_b128(dst, src);
    }
  }

  // ---- async weight-tile DMA: 2688 b128 / tile => exactly 21 per thread ----
  auto issue_tile = [&](int nt, int buf) {
    const uint32_t wbase = wbuf_b0 + (uint32_t)buf * wbufSz;
    const size_t cb = (size_t)c * T_ + (size_t)nt * 16;
#pragma unroll 1
    for (int i = 0; i < 21; ++i) {
      const int idx  = tid + 128 * i;                       // 0..2687
      const int matT = idx >= 1344;
      const int f    = matT ? idx - 1344 : idx;
      const int o    = f / 84;
      const int q    = f - o * 84;
      const float* src = (matT ? Wt : Ws) + (cb + o) * T_ + q * 4;
      const uint32_t dst = wbase +
          (uint32_t)((matT ? LDS_W : 0) + o * W_STRIDE + q * 4) * 4;
      async_b128(dst, src);
    }
  };

  issue_tile(0, 0);
  issue_tile(1, 1);

  const int brow = wave * 16 + m;

  for (int nt = 0; nt < T_ / 16; ++nt) {
    const int buf = nt & 1;
    if (nt >= 1 && nt + 1 < T_ / 16)
      issue_tile(nt + 1, (nt + 1) & 1);      // DMA next tile into idle buffer

    // Retire everything except the tile still in flight (in-order completion).
    if (nt + 1 < T_ / 16)
      asm volatile("s_wait_asynccnt 21" ::: "memory");
    else
      asm volatile("s_wait_asynccnt 0" ::: "memory");
    __syncthreads();                         // all waves' DMA for tile nt landed

    const float* wS = lds + 2 * LDS_ACT + buf * (2 * LDS_W);
    const float* wT = wS + LDS_W;

    v8f acc = {};
#pragma unroll 4
    for (int k = 0; k < T_; k += 4) {
      const int ka = k + 2 * h;     // lanes 0-15: K,K+1   lanes 16-31: K+2,K+3
      const v2f a_s = *(const v2f*)(aS + (ka >> 1) * PAIR_STRIDE + 2 * brow);
      const v2f a_t = *(const v2f*)(aT + (ka >> 1) * PAIR_STRIDE + 2 * brow);
      const v2f b_s = *(const v2f*)(wS + m * W_STRIDE + ka);
      const v2f b_t = *(const v2f*)(wT + m * W_STRIDE + ka);
      acc = __builtin_amdgcn_wmma_f32_16x16x4_f32(false, a_t, false, b_t,
                                                  (short)0, acc, false, false);
      acc = __builtin_amdgcn_wmma_f32_16x16x4_f32(false, a_s, false, b_s,
                                                  (short)0, acc, false, false);
    }

    // Epilogue: bias (column-only => one scalar per lane) + store
    const int o = nt * 16 + m;
    const float bias = bs[c * T_ + o] + bt[c * T_ + o];
#pragma unroll
    for (int r = 0; r < 8; ++r) {
      const int row = wave * 16 + r + 8 * h;            // C/D VGPR layout
      out[((size_t)row * T_ + o) * C_ + c] = acc[r] + bias;
    }
    __syncthreads();                         // done reading buf before reuse
  }
}

extern "C" void kernel_launch(void* const* d_in, const int* in_sizes, int n_in,
                              void* d_out, int out_size, void* d_ws, size_t ws_size,
                              hipStream_t stream) {
  const float* x  = (const float*)d_in[0];
  const float* Ws = (const float*)d_in[1];
  const float* bs = (const float*)d_in[2];
  const float* Wt = (const float*)d_in[3];
  const float* bt = (const float*)d_in[4];
  float* out  = (float*)d_out;
  float* s_ws = (float*)d_ws;                       // [C][T/2][B][2] seasonal
  float* t_ws = s_ws + (size_t)C_ * T_ * B_;        // [C][T/2][B][2] trend

  const int tslab = T_ / 6;                         // 56 (even)
  dim3 dgrid(C_ / 32, 6);
  decompose_kernel<<<dgrid, 256, 0, stream>>>(x, s_ws, t_ws, tslab);

  const size_t ldsBytes = (size_t)(2 * LDS_ACT + 4 * LDS_W) * sizeof(float);
  odd_gemm_kernel<<<C_, 128, ldsBytes, stream>>>(s_ws, t_ws, Ws, bs, Wt, bt, out);
}